// LogicLayer_57509612094159
// MI455X (gfx1250) — compile-verified
//
#include <hip/hip_runtime.h>
#include <math.h>

// Problem dims (fixed by the reference)
enum { BATCH = 2048, IN_DIM = 16384, OUT_DIM = 16384 };
enum { ROWS_PER_TILE = 128, BLOCK = 256 };

typedef __attribute__((ext_vector_type(2))) float v2f;
typedef __attribute__((ext_vector_type(8))) float v8f;

// _OP_COEFFS[16][4] from the reference
__device__ __constant__ float OPC[16][4] = {
    {0.f, 0.f, 0.f, 0.f}, {0.f, 0.f, 0.f, 1.f}, {0.f, 1.f, 0.f, -1.f}, {0.f, 1.f, 0.f, 0.f},
    {0.f, 0.f, 1.f, -1.f}, {0.f, 0.f, 1.f, 0.f}, {0.f, 1.f, 1.f, -2.f}, {0.f, 1.f, 1.f, -1.f},
    {1.f, -1.f, -1.f, 1.f}, {1.f, -1.f, -1.f, 2.f}, {1.f, 0.f, -1.f, 0.f}, {1.f, 0.f, -1.f, 1.f},
    {1.f, -1.f, 0.f, 0.f}, {1.f, -1.f, 0.f, 1.f}, {1.f, 0.f, 0.f, -1.f}, {1.f, 0.f, 0.f, 0.f}};

// ---------------------------------------------------------------------------
// Phase 1: coef[g][0..3] = softmax(w[g][:]) @ OPC  via V_WMMA_F32_16X16X4_F32.
// One wave32 handles 16 gates (rows M=0..15). K=16 is split into 4 chunks of 4.
// A (16x4 f32) layout per ISA 7.12.2: lane L -> M = L%16, h = L/16;
//   VGPR0 = A[M][K= 2h], VGPR1 = A[M][K= 2h+1]   (K pair per half-wave).
// B (4x16 f32): lane L -> N = L%16; VGPR0 = B[2h][N], VGPR1 = B[2h+1][N].
// D (16x16 f32): VGPR v at lane L = D[v + 8h][L%16].
// ---------------------------------------------------------------------------
__global__ void __launch_bounds__(BLOCK)
coef_wmma_kernel(const float* __restrict__ w, float* __restrict__ coef) {
    const int lane = threadIdx.x & 31;
    const int wave = threadIdx.x >> 5;
    const int base = (blockIdx.x * (BLOCK / 32) + wave) * 16;  // first gate of this wave
    const int m = lane & 15;
    const int h = lane >> 4;  // half-wave: selects K pair {2h, 2h+1}
    const float* wr = w + (size_t)(base + m) * 16;

    // This lane's 8 logits: k = 4c + 2h + {0,1}, c = 0..3 (the other half-wave
    // lane with the same m holds the complementary 8).
    float e[8];
    float mx = -INFINITY;
#pragma unroll
    for (int c = 0; c < 4; ++c) {
        e[2 * c]     = wr[4 * c + 2 * h];
        e[2 * c + 1] = wr[4 * c + 2 * h + 1];
        mx = fmaxf(mx, fmaxf(e[2 * c], e[2 * c + 1]));
    }
    mx = fmaxf(mx, __shfl_xor(mx, 16, 32));  // row max across the lane pair
    float s = 0.f;
#pragma unroll
    for (int i = 0; i < 8; ++i) { e[i] = __expf(e[i] - mx); s += e[i]; }
    s += __shfl_xor(s, 16, 32);              // row sum across the lane pair
    const float inv = 1.0f / s;

    v8f d = {};
#pragma unroll
    for (int c = 0; c < 4; ++c) {
        v2f a, b;
        a.x = e[2 * c] * inv;
        a.y = e[2 * c + 1] * inv;
        const int k0 = 4 * c + 2 * h;        // global K of this lane's B rows
        b.x = (m < 4) ? OPC[k0][m] : 0.f;    // N>=4 columns are zero padding
        b.y = (m < 4) ? OPC[k0 + 1][m] : 0.f;
        d = __builtin_amdgcn_wmma_f32_16x16x4_f32(
            /*neg_a=*/false, a, /*neg_b=*/false, b,
            /*c_mod=*/(short)0, d, /*reuse_a=*/false, /*reuse_b=*/false);
    }

    // Scatter D back: lane (m<4, h) holds coef[base + v + 8h][m] in d[v].
    if (m < 4) {
#pragma unroll
        for (int v = 0; v < 8; ++v)
            coef[(size_t)(base + v + 8 * h) * 4 + m] = d[v];
    }
}

// ---------------------------------------------------------------------------
// Phase 2: bandwidth-bound gather + fused-multiply. Thread-per-column, loop
// over a tile of rows. Gathers hit the L2-resident x (134MB < 192MB L2);
// output uses non-temporal stores so the stream does not evict x from L2.
// ---------------------------------------------------------------------------
__global__ void __launch_bounds__(BLOCK)
logic_main_kernel(const float* __restrict__ x, const int* __restrict__ idx_a,
                  const int* __restrict__ idx_b, const float4* __restrict__ coef,
                  float* __restrict__ out) {
    const int j = blockIdx.x * BLOCK + threadIdx.x;        // output column
    const int r0 = blockIdx.y * ROWS_PER_TILE;             // first row of tile
    const int ia = idx_a[j];
    const int ib = idx_b[j];
    const float4 c = coef[j];

    const float* xr = x + (size_t)r0 * IN_DIM;
    float* op = out + (size_t)r0 * OUT_DIM + j;
#pragma unroll 4
    for (int i = 0; i < ROWS_PER_TILE; ++i) {
        const float a = xr[ia];
        const float b = xr[ib];
        // c0 + c1*a + c2*b + c3*a*b  ==  a*(c3*b + c1) + (c2*b + c0)
        const float v = fmaf(a, fmaf(c.w, b, c.y), fmaf(c.z, b, c.x));
        __builtin_nontemporal_store(v, op);
        xr += IN_DIM;
        op += OUT_DIM;
    }
}

// ---------------------------------------------------------------------------
// Fallback if the workspace cannot hold the 256KB coef table: recompute the
// softmax coefficients per column inside the main kernel (scalar, cheap).
// ---------------------------------------------------------------------------
__global__ void __launch_bounds__(BLOCK)
logic_fused_kernel(const float* __restrict__ x, const int* __restrict__ idx_a,
                   const int* __restrict__ idx_b, const float* __restrict__ w,
                   float* __restrict__ out) {
    const int j = blockIdx.x * BLOCK + threadIdx.x;
    const int r0 = blockIdx.y * ROWS_PER_TILE;
    const int ia = idx_a[j];
    const int ib = idx_b[j];

    const float* wr = w + (size_t)j * 16;
    float mx = wr[0];
#pragma unroll
    for (int k = 1; k < 16; ++k) mx = fmaxf(mx, wr[k]);
    float s = 0.f, c0 = 0.f, c1 = 0.f, c2 = 0.f, c3 = 0.f;
#pragma unroll
    for (int k = 0; k < 16; ++k) {
        const float p = __expf(wr[k] - mx);
        s += p;
        c0 = fmaf(p, OPC[k][0], c0);
        c1 = fmaf(p, OPC[k][1], c1);
        c2 = fmaf(p, OPC[k][2], c2);
        c3 = fmaf(p, OPC[k][3], c3);
    }
    const float inv = 1.0f / s;
    c0 *= inv; c1 *= inv; c2 *= inv; c3 *= inv;

    const float* xr = x + (size_t)r0 * IN_DIM;
    float* op = out + (size_t)r0 * OUT_DIM + j;
#pragma unroll 4
    for (int i = 0; i < ROWS_PER_TILE; ++i) {
        const float a = xr[ia];
        const float b = xr[ib];
        const float v = fmaf(a, fmaf(c3, b, c1), fmaf(c2, b, c0));
        __builtin_nontemporal_store(v, op);
        xr += IN_DIM;
        op += OUT_DIM;
    }
}

extern "C" void kernel_launch(void* const* d_in, const int* in_sizes, int n_in,
                              void* d_out, int out_size, void* d_ws, size_t ws_size,
                              hipStream_t stream) {
    const float* x     = (const float*)d_in[0];
    const int* idx_a   = (const int*)d_in[1];
    const int* idx_b   = (const int*)d_in[2];
    const float* wgt   = (const float*)d_in[3];
    float* out         = (float*)d_out;

    const dim3 grid(OUT_DIM / BLOCK, BATCH / ROWS_PER_TILE);
    const size_t coef_bytes = (size_t)OUT_DIM * 4 * sizeof(float);

    if (ws_size >= coef_bytes) {
        float* coef = (float*)d_ws;
        // 16 gates per wave, 8 waves per block -> 128 gates/block
        coef_wmma_kernel<<<OUT_DIM / 128, BLOCK, 0, stream>>>(wgt, coef);
        logic_main_kernel<<<grid, BLOCK, 0, stream>>>(x, idx_a, idx_b,
                                                      (const float4*)coef, out);
    } else {
        logic_fused_kernel<<<grid, BLOCK, 0, stream>>>(x, idx_a, idx_b, wgt, out);
    }
}